// SAAF_11416023073153
// MI455X (gfx1250) — compile-verified
//
#include <hip/hip_runtime.h>

// ---------------------------------------------------------------------------
// SAAF elementwise kernel, MI455X (gfx1250).
// Memory-bound: 16MB x in + 16MB out; floor ~1.4us @ 23.3 TB/s.
// Coefficients (2.5MB, L2-resident) staged to LDS via gfx1250 async
// global->LDS DMA (GLOBAL_LOAD_ASYNC_TO_LDS_B32, per-lane addresses do the
// t<->c transpose in flight), then register-cached across 4 batch slices.
// Staging is straight-line (fully unrolled, branch-free) to avoid divergent
// exec-mask loops.
// ---------------------------------------------------------------------------

#define T_DIM 4096
#define C_DIM 32
#define B_DIM 32
#define TT    32            // t-values per block tile
#define NB    4             // batch slices per block
#define LDS_STRIDE 36       // padded c-stride (multiple of 4, avoids bank clash)

#if defined(__has_builtin)
#  if __has_builtin(__builtin_amdgcn_global_load_async_to_lds_b32)
#    define SAAF_ASYNC 1
#  endif
#endif

#ifdef SAAF_ASYNC
typedef __attribute__((address_space(1))) int GInt;   // global (AS1) int
typedef __attribute__((address_space(3))) int LInt;   // LDS (AS3) int
__device__ __forceinline__ void saaf_stage_b32(const float* gp, float* lp) {
  // GLOBAL_LOAD_ASYNC_TO_LDS_B32: per-lane global addr -> per-lane LDS addr
  __builtin_amdgcn_global_load_async_to_lds_b32((GInt*)gp, (LInt*)lp, 0, 0);
}
#else
__device__ __forceinline__ void saaf_stage_b32(const float* gp, float* lp) {
  *lp = *gp;
}
#endif

__device__ __forceinline__ float saaf_eval(float x, float v0, float v1,
                                           float w0, float w1, float w2) {
  // Breakpoints with the reference's exact f32 rounding:
  // bp = arange(4) * f32(4096/3); note 3*BP1 rounds to 4096.0 -> DK2 != DK0.
  constexpr float BP1 = (float)(4096.0 / 3.0);   // 1365.3333740234375
  constexpr float BP2 = 2.0f * BP1;              // exact
  constexpr float BP3 = 3.0f * BP1;              // rounds to 4096.0
  constexpr float DK0 = BP1 - 0.0f;
  constexpr float DK1 = BP2 - BP1;               // == BP1
  constexpr float DK2 = BP3 - BP2;               // 1365.333251953125
  constexpr float H0  = 0.5f * DK0 * DK0;        // (0.5*dk)*dk as in reference
  constexpr float H1  = 0.5f * DK1 * DK1;
  constexpr float H2  = 0.5f * DK2 * DK2;

  float r = v0 + x * v1;                         // sigma1 = v0 + x*v1

  float d0 = x - 0.0f;
  float b0 = (x > 0.0f && x < BP1) ? (0.5f * (d0 * d0))
                                   : (H0 + DK0 * (x - BP1));
  float d1 = x - BP1;
  float b1 = (x > BP1 && x < BP2)  ? (0.5f * (d1 * d1))
                                   : (H1 + DK1 * (x - BP2));
  float d2 = x - BP2;
  float b2 = (x > BP2 && x < BP3)  ? (0.5f * (d2 * d2))
                                   : (H2 + DK2 * (x - BP3));

  r += b0 * w0;
  r += b1 * w1;
  r += b2 * w2;
  return r;
}

__global__ __launch_bounds__(256) void saaf_kernel(
    const float* __restrict__ x,
    const float* __restrict__ v,
    const float* __restrict__ w,
    float* __restrict__ out)
{
  // sC[k][t_local*36 + c]; k: 0..1 -> v(j=0,1), 2..4 -> w(k=0..2).  23.0 KB.
  __shared__ float sC[5][TT * LDS_STRIDE];

  const int tid  = threadIdx.x;
  const int lane = tid & 31;          // t_local for the staging loads
  const int wv   = tid >> 5;          // wave id 0..7
  const int tile = blockIdx.x & 127;  // 128 t-tiles
  const int bg   = blockIdx.x >> 7;   // 8 batch groups
  const int t0   = tile * TT;
  const int b0   = bg * NB;

  // ---- Stage coefficient tile (branch-free, fully unrolled) ----
  // v is (C,2,T): flat row index seg = c*2+j, address = v + seg*T + t.
  // w is (C,3,T): flat row index seg = c*3+k, address = w + seg*T + t.
  const float* vbase = v + (size_t)t0 + lane;
  const float* wbase = w + (size_t)t0 + lane;
  float* lrow = &sC[0][lane * LDS_STRIDE];      // this lane's t_local row

#pragma unroll
  for (int i = 0; i < 8; ++i) {                 // 64 v-rows / 8 waves
    const int seg = wv + 8 * i;                 // 0..63
    const int j   = seg & 1;
    const int c   = seg >> 1;
    saaf_stage_b32(vbase + (size_t)seg * T_DIM,
                   lrow + j * (TT * LDS_STRIDE) + c);
  }
#pragma unroll
  for (int i = 0; i < 12; ++i) {                // 96 w-rows / 8 waves
    const int seg = wv + 8 * i;                 // 0..95
    const int kk  = seg % 3;
    const int c   = seg / 3;
    saaf_stage_b32(wbase + (size_t)seg * T_DIM,
                   lrow + (2 + kk) * (TT * LDS_STRIDE) + c);
  }

#ifdef SAAF_ASYNC
#  if __has_builtin(__builtin_amdgcn_s_wait_asynccnt)
  __builtin_amdgcn_s_wait_asynccnt(0);
#  else
  asm volatile("s_wait_asynccnt 0" ::: "memory");
#  endif
#endif
  __syncthreads();

  // ---- Compute: thread owns (t_local = tid>>3, c0 = (tid&7)*4), NB batches --
  const int tl = tid >> 3;
  const int c0 = (tid & 7) << 2;

  const float4 cv0 = *(const float4*)&sC[0][tl * LDS_STRIDE + c0];
  const float4 cv1 = *(const float4*)&sC[1][tl * LDS_STRIDE + c0];
  const float4 cw0 = *(const float4*)&sC[2][tl * LDS_STRIDE + c0];
  const float4 cw1 = *(const float4*)&sC[3][tl * LDS_STRIDE + c0];
  const float4 cw2 = *(const float4*)&sC[4][tl * LDS_STRIDE + c0];

  const size_t base_tc = (size_t)(t0 + tl) * C_DIM + c0;

  float4 xv[NB];
#pragma unroll
  for (int ib = 0; ib < NB; ++ib) {
    const size_t flat = (size_t)(b0 + ib) * (T_DIM * C_DIM) + base_tc;
    xv[ib] = *(const float4*)(x + flat);
  }

#pragma unroll
  for (int ib = 0; ib < NB; ++ib) {
    float4 r;
    r.x = saaf_eval(xv[ib].x, cv0.x, cv1.x, cw0.x, cw1.x, cw2.x);
    r.y = saaf_eval(xv[ib].y, cv0.y, cv1.y, cw0.y, cw1.y, cw2.y);
    r.z = saaf_eval(xv[ib].z, cv0.z, cv1.z, cw0.z, cw1.z, cw2.z);
    r.w = saaf_eval(xv[ib].w, cv0.w, cv1.w, cw0.w, cw1.w, cw2.w);
    const size_t flat = (size_t)(b0 + ib) * (T_DIM * C_DIM) + base_tc;
    *(float4*)(out + flat) = r;
  }
}

extern "C" void kernel_launch(void* const* d_in, const int* in_sizes, int n_in,
                              void* d_out, int out_size, void* d_ws, size_t ws_size,
                              hipStream_t stream) {
  const float* x = (const float*)d_in[0];
  const float* v = (const float*)d_in[1];
  const float* w = (const float*)d_in[2];
  float* out = (float*)d_out;

  // 128 t-tiles * (32/NB) batch groups = 1024 blocks of 256 threads.
  dim3 grid(128 * (B_DIM / NB));
  saaf_kernel<<<grid, dim3(256), 0, stream>>>(x, v, w, out);
}